// Conv2d_ONI_24988119728398
// MI455X (gfx1250) — compile-verified
//
#include <hip/hip_runtime.h>
#include <hip/hip_bf16.h>

typedef __attribute__((ext_vector_type(2))) float v2f;
typedef __attribute__((ext_vector_type(4))) float v4f;
typedef __attribute__((ext_vector_type(8))) float v8f;

#define HW     16384      // 128*128
#define CIN    64
#define COUT   64
#define NIMG   32
#define WPAD   68         // LDS row pitch for weights: bank-conflict-free A fetch
#define SQRT2F 1.41421356237309504880f

// ---------------------------------------------------------------------------
// Kernel 1: ONI weight preparation (single workgroup; tiny 64x64 matmuls).
//   zc = z - rowmean(z)
//   m  = zc zc^T ; v = zc * rsqrt(||m||_F) ; s = m / ||m||_F  (== v v^T)
//   b  = (3I - s)/2 ; 4x: b = 1.5 b - 0.5 * b@(b@(b@s))
//   w  = (b^T v) * g * sqrt(2)
// ---------------------------------------------------------------------------
__device__ __forceinline__ void mm64(float* __restrict__ D,
                                     const float* __restrict__ A,
                                     const float* __restrict__ B,
                                     int tid, int nth) {
  for (int i = tid; i < 4096; i += nth) {
    const int r = i >> 6, c = i & 63;
    float s = 0.f;
#pragma unroll
    for (int k = 0; k < 64; ++k) s = fmaf(A[(r << 6) + k], B[(k << 6) + c], s);
    D[i] = s;
  }
}

__global__ __launch_bounds__(1024) void oni_weight_kernel(
    const float* __restrict__ z, const float* __restrict__ g,
    float* __restrict__ w_out, float* __restrict__ v_glob) {
  __shared__ float S[4096];   // s
  __shared__ float Bm[4096];  // b
  __shared__ float T[4096];   // temp
  __shared__ float U[4096];   // zc / v / temp (v round-trips via v_glob)
  __shared__ float s_sumsq;

  const int tid = threadIdx.x;
  const int nth = blockDim.x;
  if (tid == 0) s_sumsq = 0.f;
  for (int i = tid; i < 4096; i += nth) U[i] = z[i];
  __syncthreads();

  if (tid < 64) {                          // row-center (z @ (I - 11^T/64))
    float s = 0.f;
    for (int k = 0; k < 64; ++k) s += U[(tid << 6) + k];
    const float mu = s * (1.f / 64.f);
    for (int k = 0; k < 64; ++k) U[(tid << 6) + k] -= mu;
  }
  __syncthreads();

  float lsq = 0.f;                         // m = zc zc^T, Frobenius^2 of m
  for (int i = tid; i < 4096; i += nth) {
    const int r = i >> 6, c = i & 63;
    float s = 0.f;
    for (int k = 0; k < 64; ++k) s = fmaf(U[(r << 6) + k], U[(c << 6) + k], s);
    S[i] = s;
    lsq = fmaf(s, s, lsq);
  }
  atomicAdd(&s_sumsq, lsq);                // ds_add_f32 on LDS
  __syncthreads();

  const float inv_norm = rsqrtf(s_sumsq);          // 1/||m||_F
  const float vscale   = rsqrtf(sqrtf(s_sumsq));   // 1/sqrt(||m||_F)
  for (int i = tid; i < 4096; i += nth) {
    S[i] *= inv_norm;                      // s = v v^T
    const float vv = U[i] * vscale;        // v
    U[i] = vv;
    v_glob[i] = vv;                        // stash v (U gets clobbered below)
  }
  __syncthreads();

  for (int i = tid; i < 4096; i += nth) {
    const int r = i >> 6, c = i & 63;
    Bm[i] = (((r == c) ? 3.f : 0.f) - S[i]) * 0.5f;
  }
  __syncthreads();

  for (int it = 1; it < 5; ++it) {         // (b@b)@(b@s) == b@(b@(b@s))
    mm64(T, Bm, S, tid, nth);  __syncthreads();
    mm64(U, Bm, T, tid, nth);  __syncthreads();
    mm64(T, Bm, U, tid, nth);  __syncthreads();
    for (int i = tid; i < 4096; i += nth) Bm[i] = 1.5f * Bm[i] - 0.5f * T[i];
    __syncthreads();
  }

  for (int i = tid; i < 4096; i += nth) U[i] = v_glob[i];   // reload v
  __syncthreads();

  for (int i = tid; i < 4096; i += nth) {  // w = (b^T v) * g * sqrt2
    const int o = i >> 6, c = i & 63;
    float s = 0.f;
    for (int k = 0; k < 64; ++k) s = fmaf(Bm[(k << 6) + o], U[(k << 6) + c], s);
    w_out[i] = s * g[o] * SQRT2F;
  }
}

// ---------------------------------------------------------------------------
// Kernel 2: 1x1 conv as WMMA f32 GEMM. One wave: 16 pixels (N) x 64 out-ch (M),
// K = 64 in-ch in 16 steps of 4 -> 64 v_wmma_f32_16x16x4_f32 per wave.
// x is read exactly once, out written exactly once -> HBM-roofline optimal.
// ---------------------------------------------------------------------------
__global__ __launch_bounds__(256) void conv1x1_oni_wmma_kernel(
    const float* __restrict__ x, const float* __restrict__ w,
    const float* __restrict__ bias, float* __restrict__ out) {
  __shared__ float sW[COUT * WPAD];        // 64 x 68 padded (17408 B)
  for (int i = threadIdx.x; i < COUT * CIN; i += 256)
    sW[(i >> 6) * WPAD + (i & 63)] = w[i];
  __syncthreads();

  const int lane = threadIdx.x & 31;
  const int wave = threadIdx.x >> 5;
  const long tile = (long)blockIdx.x * 8 + wave;   // 16 pixels per wave-tile
  const long pix0 = tile << 4;
  const int  n    = (int)(pix0 >> 14);             // / HW
  const int  hw0  = (int)(pix0 & (HW - 1));
  const int  nsub = lane & 15;                     // N col (and A's M row select)
  const int  kOff = (lane >> 4) << 1;              // ISA A/B f32 layout: 0 or 2

  const float* xb = x   + ((long)n * CIN)  * HW + hw0 + nsub;
  float*       ob = out + ((long)n * COUT) * HW + hw0 + nsub;

  v8f acc0 = {}; v8f acc1 = {}; v8f acc2 = {}; v8f acc3 = {};

  for (int k = 0; k < CIN; k += 4) {
    const int r = k + kOff;
    v2f b;                                         // B(4x16): K=r,r+1 rows, N=nsub
    b.x = xb[(long)r * HW];
    b.y = xb[(long)(r + 1) * HW];
    if (k + 4 < CIN) __builtin_prefetch(&xb[(long)(r + 4) * HW], 0, 0);

    v2f a;                                         // A(16x4): M=m0+nsub, K=r,r+1
    a = *(const v2f*)&sW[( 0 + nsub) * WPAD + r];
    acc0 = __builtin_amdgcn_wmma_f32_16x16x4_f32(false, a, false, b, (short)0, acc0, false, false);
    a = *(const v2f*)&sW[(16 + nsub) * WPAD + r];
    acc1 = __builtin_amdgcn_wmma_f32_16x16x4_f32(false, a, false, b, (short)0, acc1, false, false);
    a = *(const v2f*)&sW[(32 + nsub) * WPAD + r];
    acc2 = __builtin_amdgcn_wmma_f32_16x16x4_f32(false, a, false, b, (short)0, acc2, false, false);
    a = *(const v2f*)&sW[(48 + nsub) * WPAD + r];
    acc3 = __builtin_amdgcn_wmma_f32_16x16x4_f32(false, a, false, b, (short)0, acc3, false, false);
  }

  // C/D layout: VGPR r holds M=r (lanes 0-15) / M=r+8 (lanes 16-31), N=lane&15
  const int hi = (lane >> 4) << 3;
#pragma unroll
  for (int rr = 0; rr < 8; ++rr) {
    const int m = rr + hi;
    ob[(long)( 0 + m) * HW] = acc0[rr] + bias[ 0 + m];
    ob[(long)(16 + m) * HW] = acc1[rr] + bias[16 + m];
    ob[(long)(32 + m) * HW] = acc2[rr] + bias[32 + m];
    ob[(long)(48 + m) * HW] = acc3[rr] + bias[48 + m];
  }
}

// ---------------------------------------------------------------------------
extern "C" void kernel_launch(void* const* d_in, const int* in_sizes, int n_in,
                              void* d_out, int out_size, void* d_ws, size_t ws_size,
                              hipStream_t stream) {
  (void)in_sizes; (void)n_in; (void)out_size; (void)ws_size;
  const float* x    = (const float*)d_in[0];   // [32,64,128,128]
  const float* z    = (const float*)d_in[1];   // [64,64]
  const float* g    = (const float*)d_in[2];   // [64,1]
  const float* bias = (const float*)d_in[3];   // [64]
  float* out  = (float*)d_out;
  float* w_ws = (float*)d_ws;                  // 4096 floats: orthonormal weight
  float* v_ws = w_ws + 4096;                   // 4096 floats: v stash

  oni_weight_kernel<<<1, 1024, 0, stream>>>(z, g, w_ws, v_ws);

  const int tiles  = (NIMG * HW) / 16;         // 32768 wave-tiles
  const int blocks = tiles / 8;                // 8 waves (256 thr) per block
  conv1x1_oni_wmma_kernel<<<blocks, 256, 0, stream>>>(x, w_ws, bias, out);
}